// GCNNModel_39006892982336
// MI455X (gfx1250) — compile-verified
//
#include <hip/hip_runtime.h>
#include <hip/hip_bf16.h>

typedef __attribute__((ext_vector_type(2))) float v2f;
typedef __attribute__((ext_vector_type(4))) float v4f;
typedef __attribute__((ext_vector_type(8))) float v8f;

#define IN_DIM_C 32
#define HID_C    64

// ---------------------------------------------------------------- utilities
__global__ void k_fill(float* __restrict__ p, float v, int n) {
    int i = blockIdx.x * blockDim.x + threadIdx.x;
    if (i < n) p[i] = v;
}

__global__ void k_copy(float* __restrict__ dst, const float* __restrict__ src, int n) {
    int i = blockIdx.x * blockDim.x + threadIdx.x;
    if (i < n) dst[i] = src[i];
}

// ---------------------------------------------------------------- degree / norm
__global__ void k_deg_accum(const int* __restrict__ dst, float* __restrict__ deg, int e) {
    int i = blockIdx.x * blockDim.x + threadIdx.x;
    if (i < e) atomicAdd(&deg[dst[i]], 1.0f);
}

__global__ void k_rsqrt(float* __restrict__ p, int n) {
    int i = blockIdx.x * blockDim.x + threadIdx.x;
    if (i < n) p[i] = rsqrtf(p[i]);   // deg >= 1 always (self loop)
}

// ---------------------------------------------------------------- fp32 WMMA GEMM
// H[N x 64] = A[N x K] @ W[64 x K]^T  using V_WMMA_F32_16X16X4_F32.
// Block = 128 threads (4 waves). Block b: rows 16b..16b+15. Wave w: cols 16w..16w+15.
// TAIL=false when nrows % 16 == 0 (no guards -> straight-line stores).
template<int K, bool TAIL>
__global__ void k_gemm_wmma(const float* __restrict__ A,
                            const float* __restrict__ W,
                            float* __restrict__ H, int nrows) {
    const int lane = threadIdx.x & 31;
    const int wave = threadIdx.x >> 5;
    const int hi   = lane >> 4;      // 0: K pair {0,1}; 1: K pair {2,3}
    const int l15  = lane & 15;

    int arow = blockIdx.x * 16 + l15;            // A fragment: M = lane&15
    if (TAIL && arow >= nrows) arow = nrows - 1; // clamp loads, keep EXEC all-1s
    const int bcol = wave * 16 + l15;            // B fragment: N = lane&15

    const float* ap = A + (size_t)arow * K;
    const float* wp = W + (size_t)bcol * K;

    v8f c = {};
#pragma unroll
    for (int k0 = 0; k0 < K; k0 += 4) {
        v2f a, b;
        a.x = ap[k0 + 2 * hi];
        a.y = ap[k0 + 2 * hi + 1];
        b.x = wp[k0 + 2 * hi];                   // B[k][n] = W[n][k]
        b.y = wp[k0 + 2 * hi + 1];
        c = __builtin_amdgcn_wmma_f32_16x16x4_f32(false, a, false, b,
                                                  (short)0, c, false, false);
    }
    // C/D layout: VGPR r -> row (r + 8*hi), col = lane&15
    float* hp = H + (size_t)(blockIdx.x * 16 + 8 * hi) * HID_C + wave * 16 + l15;
#pragma unroll
    for (int r = 0; r < 8; ++r) {
        if (!TAIL || (blockIdx.x * 16 + r + 8 * hi) < nrows)
            hp[(size_t)r * HID_C] = c[r];
    }
}

// ---------------------------------------------------------------- edge scatter
// One wave per edge: AGG[d][f] += H[s][f] * dinv[s]*dinv[d], 2 feats per lane.
__global__ void k_scatter(const int* __restrict__ src, const int* __restrict__ dst,
                          const float* __restrict__ dinv, const float* __restrict__ H,
                          float* __restrict__ AGG, int e) {
    int eid = blockIdx.x * (blockDim.x >> 5) + (threadIdx.x >> 5);
    if (eid >= e) return;
    int lane = threadIdx.x & 31;
    int s = src[eid], d = dst[eid];
    float nrm = dinv[s] * dinv[d];
    const float* hp = H + (size_t)s * HID_C + lane * 2;
    float* op = AGG + (size_t)d * HID_C + lane * 2;
    atomicAdd(op,     hp[0] * nrm);
    atomicAdd(op + 1, hp[1] * nrm);
}

// ---------------------------------------------------------------- finalize conv
// OUT = act( AGG + H*dinv^2 + bias )
__global__ void k_finalize(const float* __restrict__ AGG, const float* __restrict__ H,
                           const float* __restrict__ dinv, const float* __restrict__ bias,
                           float* __restrict__ OUT, int nnodes, int clip) {
    int i = blockIdx.x * blockDim.x + threadIdx.x;
    if (i >= nnodes * HID_C) return;
    int node = i >> 6, f = i & (HID_C - 1);
    float di = dinv[node];
    float v = AGG[i] + H[i] * di * di + bias[f];
    v = fmaxf(v, 0.0f);
    if (clip) v = fminf(v, 10.0f);
    OUT[i] = v;
}

// ---------------------------------------------------------------- weather MLP
__global__ void k_weather(const float* __restrict__ rain, const float* __restrict__ fut,
                          const float* __restrict__ wf1W, const float* __restrict__ wf1b,
                          const float* __restrict__ wf2W, const float* __restrict__ wf2b,
                          float* __restrict__ wfeat, int seq) {
    __shared__ float w[HID_C];
    int t = threadIdx.x;
    float acc = wf1b[t];
    const float* row = wf1W + (size_t)t * (seq + 1);
    for (int i = 0; i < seq; ++i) acc += row[i] * rain[i];
    acc += row[seq] * fut[0];
    w[t] = fmaxf(acc, 0.0f);
    __syncthreads();
    float acc2 = wf2b[t];
    const float* row2 = wf2W + (size_t)t * HID_C;
    for (int i = 0; i < HID_C; ++i) acc2 += row2[i] * w[i];
    wfeat[t] = acc2;
}

// ---------------------------------------------------------------- fc1 (HBM-bound)
// oacc[j] += sum over chunk of of1W[j][:] . combined[:], in float4 units.
// of1W streamed once (327 MB) -> non-temporal b128 loads.
__global__ void k_fc1(const float* __restrict__ of1W, const float* __restrict__ nodeF,
                      const float* __restrict__ wfeat, float* __restrict__ oacc,
                      int nf4, int total4, int chunk4) {
    const int j = blockIdx.y;
    int base = blockIdx.x * chunk4;
    int end  = base + chunk4;
    if (end > total4) end = total4;
    const v4f* wrow = (const v4f*)(of1W + (size_t)j * (size_t)total4 * 4);
    const v4f* nf   = (const v4f*)nodeF;
    const v4f* wf   = (const v4f*)wfeat;
    float acc = 0.0f;
    for (int i = base + threadIdx.x; i < end; i += blockDim.x) {
        v4f w4 = __builtin_nontemporal_load(&wrow[i]);
        v4f c4 = (i < nf4) ? nf[i] : wf[i - nf4];
        acc += w4.x * c4.x + w4.y * c4.y + w4.z * c4.z + w4.w * c4.w;
    }
    __shared__ float sm[256];
    sm[threadIdx.x] = acc;
    __syncthreads();
    for (int s = 128; s > 0; s >>= 1) {
        if (threadIdx.x < s) sm[threadIdx.x] += sm[threadIdx.x + s];
        __syncthreads();
    }
    if (threadIdx.x == 0) atomicAdd(&oacc[j], sm[0]);
}

__global__ void k_relu(const float* __restrict__ in, float* __restrict__ out, int n) {
    int i = blockIdx.x * blockDim.x + threadIdx.x;
    if (i < n) out[i] = fmaxf(in[i], 0.0f);
}

// ---------------------------------------------------------------- fc2: warp per row
// Each lane: one b128 load (4 floats) of the 128-wide row, then wave reduce.
__global__ void k_fc2(const float* __restrict__ of2W, const float* __restrict__ of2b,
                      const float* __restrict__ o, float* __restrict__ out, int nrows) {
    int w = blockIdx.x * (blockDim.x >> 5) + (threadIdx.x >> 5);
    if (w >= nrows) return;
    int lane = threadIdx.x & 31;
    const v4f* row = (const v4f*)(of2W + (size_t)w * 128);
    const v4f* o4  = (const v4f*)o;
    v4f r = __builtin_nontemporal_load(&row[lane]);
    v4f v = o4[lane];
    float acc = r.x * v.x + r.y * v.y + r.z * v.z + r.w * v.w;
#pragma unroll
    for (int off = 16; off > 0; off >>= 1) acc += __shfl_down(acc, off, 32);
    if (lane == 0) out[w] = acc + of2b[w];
}

// ---------------------------------------------------------------- host
extern "C" void kernel_launch(void* const* d_in, const int* in_sizes, int n_in,
                              void* d_out, int out_size, void* d_ws, size_t ws_size,
                              hipStream_t stream) {
    const float* x     = (const float*)d_in[0];
    const int*   ei    = (const int*)d_in[1];
    const float* rain  = (const float*)d_in[2];
    const float* fut   = (const float*)d_in[3];
    const float* W1    = (const float*)d_in[4];
    const float* b1    = (const float*)d_in[5];
    const float* W2    = (const float*)d_in[6];
    const float* b2    = (const float*)d_in[7];
    const float* wf1W  = (const float*)d_in[8];
    const float* wf1b  = (const float*)d_in[9];
    const float* wf2W  = (const float*)d_in[10];
    const float* wf2b  = (const float*)d_in[11];
    const float* of1W  = (const float*)d_in[12];
    const float* of1b  = (const float*)d_in[13];
    const float* of2W  = (const float*)d_in[14];
    const float* of2b  = (const float*)d_in[15];
    float* out = (float*)d_out;

    const int N   = in_sizes[0] / IN_DIM_C;   // 10000
    const int E   = in_sizes[1] / 2;          // 320000
    const int SEQ = in_sizes[2];              // 168
    const int NF  = N * HID_C;                // 640000

    const int* src = ei;
    const int* dst = ei + E;

    // workspace layout (floats)
    float* ws    = (float*)d_ws;
    float* dinv  = ws;                 // N
    float* h_lin = ws + 10240;         // NF
    float* agg   = h_lin + NF;         // NF
    float* h_act = agg + NF;           // NF
    float* wfeat = h_act + NF;         // 64
    float* oacc  = wfeat + 64;         // 128
    float* ovec  = oacc + 128;         // 128

    const int T = 256;
    const int gN  = (N + T - 1) / T;
    const int gE  = (E + T - 1) / T;
    const int gNF = (NF + T - 1) / T;
    const int gTile = (N + 15) / 16;          // 625
    const int gEdgeWaves = (E + 7) / 8;       // 8 waves (edges) per 256-thr block
    const bool even = (N % 16) == 0;

    // ---- degrees / normalization
    k_fill<<<gN, T, 0, stream>>>(dinv, 1.0f, N);               // self-loop
    k_deg_accum<<<gE, T, 0, stream>>>(dst, dinv, E);
    k_rsqrt<<<gN, T, 0, stream>>>(dinv, N);

    // ---- GCN layer 1
    if (even) k_gemm_wmma<IN_DIM_C, false><<<gTile, 128, 0, stream>>>(x, W1, h_lin, N);
    else      k_gemm_wmma<IN_DIM_C, true ><<<gTile, 128, 0, stream>>>(x, W1, h_lin, N);
    k_fill<<<gNF, T, 0, stream>>>(agg, 0.0f, NF);
    k_scatter<<<gEdgeWaves, T, 0, stream>>>(src, dst, dinv, h_lin, agg, E);
    k_finalize<<<gNF, T, 0, stream>>>(agg, h_lin, dinv, b1, h_act, N, 0);

    // ---- GCN layer 2
    if (even) k_gemm_wmma<HID_C, false><<<gTile, 128, 0, stream>>>(h_act, W2, h_lin, N);
    else      k_gemm_wmma<HID_C, true ><<<gTile, 128, 0, stream>>>(h_act, W2, h_lin, N);
    k_fill<<<gNF, T, 0, stream>>>(agg, 0.0f, NF);
    k_scatter<<<gEdgeWaves, T, 0, stream>>>(src, dst, dinv, h_lin, agg, E);
    k_finalize<<<gNF, T, 0, stream>>>(agg, h_lin, dinv, b2, h_act, N, 1);

    // ---- weather path
    k_weather<<<1, HID_C, 0, stream>>>(rain, fut, wf1W, wf1b, wf2W, wf2b, wfeat, SEQ);

    // ---- fc1: 128-row x 640064 GEMV, split along K across blocks (float4 units)
    k_copy<<<1, 128, 0, stream>>>(oacc, of1b, 128);
    const int total4 = (NF + HID_C) / 4;                       // 160016
    const int nf4    = NF / 4;                                 // 160000
    const int CHUNK4 = 4096;                                   // 16384 floats
    const int nchunk = (total4 + CHUNK4 - 1) / CHUNK4;         // 40
    k_fc1<<<dim3(nchunk, 128), T, 0, stream>>>(of1W, h_act, wfeat, oacc,
                                               nf4, total4, CHUNK4);
    k_relu<<<1, 128, 0, stream>>>(oacc, ovec, 128);

    // ---- fc2: 10000 x 128 GEMV, warp per row
    const int gOut = (N + 7) / 8;
    k_fc2<<<gOut, T, 0, stream>>>(of2W, of2b, ovec, out, N);
}